// GATLayer_3539053052701
// MI455X (gfx1250) — compile-verified
//
#include <hip/hip_runtime.h>
#include <hip/hip_bf16.h>

#define IN_DIM   128
#define OUT_DIM  64
#define HEADS    4
#define CH       (HEADS * OUT_DIM)   // 256
#define NEG_SLOPE 0.2f
#define LN_EPS    1e-5f

typedef __attribute__((ext_vector_type(2))) float v2f;
typedef __attribute__((ext_vector_type(8))) float v8f;

// ---------------------------------------------------------------------------
// Init: zero the scatter accumulator (d_out), set segment-max to -inf, sums 0
// ---------------------------------------------------------------------------
__global__ void gat_init(float* __restrict__ out, float* __restrict__ mbuf,
                         float* __restrict__ sbuf, int N) {
    const int stride = gridDim.x * blockDim.x;
    const int tid = blockIdx.x * blockDim.x + threadIdx.x;
    const int total = N * CH;
    for (int i = tid; i < total; i += stride) out[i] = 0.0f;
    const int nh = N * HEADS;
    const float ninf = __int_as_float(0xFF800000);  // -inf
    for (int i = tid; i < nh; i += stride) { mbuf[i] = ninf; sbuf[i] = 0.0f; }
}

// ---------------------------------------------------------------------------
// GEMM: xw[N,256] = x[N,128] @ W[128,256] via V_WMMA_F32_16X16X4_F32
// One wave per 16x16 output tile; K=128 fully unrolled (32 WMMAs/wave).
// A 16x4 f32 layout: lanes0-15 hold K=k0,k0+1; lanes16-31 hold K=k0+2,k0+3.
// B 4x16 f32 layout mirrors A with N striped across lanes.
// C/D: VGPR r -> row M=r (lanes0-15) / M=r+8 (lanes16-31), col = lane%16.
// ---------------------------------------------------------------------------
__global__ void gat_gemm_wmma(const float* __restrict__ x,
                              const float* __restrict__ W,
                              float* __restrict__ xw, int N) {
    const int wave = blockIdx.x * (blockDim.x >> 5) + (threadIdx.x >> 5);
    const int rowTile = wave >> 4;          // 16 column tiles (256/16)
    const int colTile = wave & 15;
    if (rowTile * 16 >= N) return;          // wave-uniform exit
    const int lane = threadIdx.x & 31;
    const int half = lane >> 4;             // 0 or 1
    const int l16  = lane & 15;
    const int row  = rowTile * 16 + l16;
    const int col  = colTile * 16 + l16;
    const float* __restrict__ xrow = x + (size_t)row * IN_DIM;

    v8f acc = {};
#pragma unroll
    for (int k0 = 0; k0 < IN_DIM; k0 += 4) {
        const int ka = k0 + 2 * half;
        v2f a;
        a.x = xrow[ka];
        a.y = xrow[ka + 1];
        v2f b;
        b.x = W[(size_t)ka * CH + col];
        b.y = W[(size_t)(ka + 1) * CH + col];
        acc = __builtin_amdgcn_wmma_f32_16x16x4_f32(
            /*neg_a=*/false, a, /*neg_b=*/false, b,
            /*c_mod=*/(short)0, acc, /*reuse_a=*/false, /*reuse_b=*/false);
    }
#pragma unroll
    for (int r = 0; r < 8; ++r) {
        const int m = r + half * 8;
        xw[(size_t)(rowTile * 16 + m) * CH + colTile * 16 + l16] = acc[r];
    }
}

// ---------------------------------------------------------------------------
// Per-node attention logits: a_s[n,h] = <xw[n,h,:], att_src[h,:]>, same a_d
// ---------------------------------------------------------------------------
__global__ void gat_logits(const float* __restrict__ xw,
                           const float* __restrict__ att_src,
                           const float* __restrict__ att_dst,
                           float* __restrict__ a_s, float* __restrict__ a_d,
                           int N) {
    const int idx = blockIdx.x * blockDim.x + threadIdx.x;  // n*H + h
    if (idx >= N * HEADS) return;
    const int n = idx >> 2, h = idx & 3;
    const float* __restrict__ v  = xw + (size_t)n * CH + h * OUT_DIM;
    const float* __restrict__ as = att_src + h * OUT_DIM;
    const float* __restrict__ ad = att_dst + h * OUT_DIM;
    float ss = 0.f, sd = 0.f;
#pragma unroll 8
    for (int d = 0; d < OUT_DIM; ++d) {
        const float t = v[d];
        ss += t * as[d];
        sd += t * ad[d];
    }
    a_s[idx] = ss;
    a_d[idx] = sd;
}

// ---------------------------------------------------------------------------
// Edge helpers. Self loops: edge e in [E, E+N) has src = dst = e - E.
// edge_index is int64 [2,E]: row 0 = src, row 1 = dst.
// ---------------------------------------------------------------------------
__device__ __forceinline__ void edge_sd(const long long* __restrict__ ei,
                                        int e, int E, int& s, int& d) {
    if (e < E) { s = (int)ei[e]; d = (int)ei[(size_t)E + e]; }
    else       { s = d = e - E; }
}

__device__ __forceinline__ float leaky(float v) {
    return v > 0.0f ? v : NEG_SLOPE * v;
}

// Sign-partitioned atomic float max (init must be -inf).
__device__ __forceinline__ void atomicMaxF(float* addr, float val) {
    if (val >= 0.0f) atomicMax((int*)addr, __float_as_int(val));
    else             atomicMin((unsigned int*)addr, __float_as_uint(val));
}

// ---------------------------------------------------------------------------
// Segment max of leaky-relu edge scores over incoming edges of dst
// ---------------------------------------------------------------------------
__global__ void gat_edge_max(const long long* __restrict__ ei,
                             const float* __restrict__ a_s,
                             const float* __restrict__ a_d,
                             float* __restrict__ mbuf, int E, int N) {
    const int idx = blockIdx.x * blockDim.x + threadIdx.x;  // e*H + h
    const int Et = E + N;
    if (idx >= Et * HEADS) return;
    const int e = idx >> 2, h = idx & 3;
    int s, d;
    edge_sd(ei, e, E, s, d);
    const float v = leaky(a_s[s * HEADS + h] + a_d[d * HEADS + h]);
    atomicMaxF(&mbuf[d * HEADS + h], v);
}

// ---------------------------------------------------------------------------
// Segment sum of exp(e - max)
// ---------------------------------------------------------------------------
__global__ void gat_edge_sum(const long long* __restrict__ ei,
                             const float* __restrict__ a_s,
                             const float* __restrict__ a_d,
                             const float* __restrict__ mbuf,
                             float* __restrict__ sbuf, int E, int N) {
    const int idx = blockIdx.x * blockDim.x + threadIdx.x;
    const int Et = E + N;
    if (idx >= Et * HEADS) return;
    const int e = idx >> 2, h = idx & 3;
    int s, d;
    edge_sd(ei, e, E, s, d);
    const float v  = leaky(a_s[s * HEADS + h] + a_d[d * HEADS + h]);
    const float ex = __expf(v - mbuf[d * HEADS + h]);
    atomicAdd(&sbuf[d * HEADS + h], ex);
}

// ---------------------------------------------------------------------------
// Scatter-aggregate: out[dst] += alpha * xw[src]. One wave32 per edge,
// each lane owns 8 contiguous channels (single head per lane). Alpha is
// recomputed per lane (cheap; avoids a 13.6 MB per-edge buffer).
// ---------------------------------------------------------------------------
__global__ void gat_aggregate(const long long* __restrict__ ei,
                              const float* __restrict__ a_s,
                              const float* __restrict__ a_d,
                              const float* __restrict__ mbuf,
                              const float* __restrict__ sbuf,
                              const float* __restrict__ xw,
                              float* __restrict__ out, int E, int N) {
    const int wave = blockIdx.x * (blockDim.x >> 5) + (threadIdx.x >> 5);
    const int Et = E + N;
    if (wave >= Et) return;
    const int lane = threadIdx.x & 31;
    int s, d;
    edge_sd(ei, wave, E, s, d);
    const int c0 = lane * 8;        // channel range [c0, c0+8) -> one head
    const int h  = c0 >> 6;
    const float v = leaky(a_s[s * HEADS + h] + a_d[d * HEADS + h]);
    const float alpha = __expf(v - mbuf[d * HEADS + h]) /
                        (sbuf[d * HEADS + h] + 1e-16f);
    const float4* __restrict__ src4 = (const float4*)(xw + (size_t)s * CH + c0);
    float* __restrict__ dst = out + (size_t)d * CH + c0;
    const float4 va = src4[0];
    const float4 vb = src4[1];
    atomicAdd(dst + 0, alpha * va.x);
    atomicAdd(dst + 1, alpha * va.y);
    atomicAdd(dst + 2, alpha * va.z);
    atomicAdd(dst + 3, alpha * va.w);
    atomicAdd(dst + 4, alpha * vb.x);
    atomicAdd(dst + 5, alpha * vb.y);
    atomicAdd(dst + 6, alpha * vb.z);
    atomicAdd(dst + 7, alpha * vb.w);
}

// ---------------------------------------------------------------------------
// In-place bias + LayerNorm over 256 channels. One block per node.
// ---------------------------------------------------------------------------
__global__ void gat_layernorm(float* __restrict__ out,
                              const float* __restrict__ bias,
                              const float* __restrict__ gamma,
                              const float* __restrict__ beta, int N) {
    __shared__ float red[CH];
    const int n = blockIdx.x;
    const int c = threadIdx.x;
    float v = out[(size_t)n * CH + c] + bias[c];

    red[c] = v;
    __syncthreads();
#pragma unroll
    for (int off = CH / 2; off > 0; off >>= 1) {
        if (c < off) red[c] += red[c + off];
        __syncthreads();
    }
    const float mu = red[0] * (1.0f / CH);
    __syncthreads();

    const float dv = v - mu;
    red[c] = dv * dv;
    __syncthreads();
#pragma unroll
    for (int off = CH / 2; off > 0; off >>= 1) {
        if (c < off) red[c] += red[c + off];
        __syncthreads();
    }
    const float var = red[0] * (1.0f / CH);
    const float r = rsqrtf(var + LN_EPS);
    out[(size_t)n * CH + c] = dv * r * gamma[c] + beta[c];
}

// ---------------------------------------------------------------------------
// Launch
// ---------------------------------------------------------------------------
extern "C" void kernel_launch(void* const* d_in, const int* in_sizes, int n_in,
                              void* d_out, int out_size, void* d_ws, size_t ws_size,
                              hipStream_t stream) {
    const float*     x        = (const float*)d_in[0];
    const long long* ei       = (const long long*)d_in[1];   // int64 [2,E]
    const float*     W        = (const float*)d_in[2];
    const float*     att_src  = (const float*)d_in[3];
    const float*     att_dst  = (const float*)d_in[4];
    const float*     bias     = (const float*)d_in[5];
    const float*     gamma    = (const float*)d_in[6];
    const float*     beta     = (const float*)d_in[7];
    float*           out      = (float*)d_out;

    const int N  = in_sizes[0] / IN_DIM;
    const int E  = in_sizes[1] / 2;
    const int Et = E + N;

    // Workspace layout (floats): xw[N*256] | a_s[N*4] | a_d[N*4] | m[N*4] | s[N*4]
    float* ws   = (float*)d_ws;
    float* xw   = ws;
    float* a_s  = xw  + (size_t)N * CH;
    float* a_d  = a_s + (size_t)N * HEADS;
    float* mbuf = a_d + (size_t)N * HEADS;
    float* sbuf = mbuf + (size_t)N * HEADS;

    gat_init<<<1024, 256, 0, stream>>>(out, mbuf, sbuf, N);

    const int rowTiles = (N + 15) / 16;
    const int waves = rowTiles * 16;                 // 16 column tiles of 256
    gat_gemm_wmma<<<(waves + 7) / 8, 256, 0, stream>>>(x, W, xw, N);

    gat_logits<<<(N * HEADS + 255) / 256, 256, 0, stream>>>(xw, att_src, att_dst,
                                                            a_s, a_d, N);
    gat_edge_max<<<(Et * HEADS + 255) / 256, 256, 0, stream>>>(ei, a_s, a_d,
                                                               mbuf, E, N);
    gat_edge_sum<<<(Et * HEADS + 255) / 256, 256, 0, stream>>>(ei, a_s, a_d,
                                                               mbuf, sbuf, E, N);
    gat_aggregate<<<(Et + 7) / 8, 256, 0, stream>>>(ei, a_s, a_d, mbuf, sbuf,
                                                    xw, out, E, N);
    gat_layernorm<<<N, CH, 0, stream>>>(out, bias, gamma, beta, N);
}